// LossRNNT_21492016349393
// MI455X (gfx1250) — compile-verified
//
#include <hip/hip_runtime.h>
#include <cstdint>
#include <cstddef>

#define NEGF (-1e30f)

constexpr int B_ = 8, T_ = 256, U_ = 64, U1_ = 65, V_ = 1024;

typedef float v4f __attribute__((ext_vector_type(4)));

// ---------------------------------------------------------------------------
// Pass 1: per-(b,t,u) row logsumexp over V=1024; emit blank/emit log-probs.
// One wave32 per row, 8 waves per block. Pure streaming: 8x b128 (NT) per lane.
// ---------------------------------------------------------------------------
__global__ __launch_bounds__(256)
void rnnt_logprobs_kernel(const float* __restrict__ logits,
                          const int* __restrict__ labels,
                          const int* __restrict__ y_len,
                          float* __restrict__ blank_ws,
                          float* __restrict__ emit_ws) {
  const int lane = threadIdx.x & 31;
  const int wave = threadIdx.x >> 5;
  const int row  = blockIdx.x * 8 + wave;          // in [0, B*T*U1), exact fit
  const v4f* __restrict__ x4 = (const v4f*)(logits + (size_t)row * V_);

  v4f d[8];
#pragma unroll
  for (int i = 0; i < 8; ++i)
    d[i] = __builtin_nontemporal_load(x4 + i * 32 + lane);  // b128, TH=NT

  float m = -3.402823466e38f;
#pragma unroll
  for (int i = 0; i < 8; ++i)
    m = fmaxf(m, fmaxf(fmaxf(d[i][0], d[i][1]), fmaxf(d[i][2], d[i][3])));
#pragma unroll
  for (int off = 16; off > 0; off >>= 1) m = fmaxf(m, __shfl_xor(m, off, 32));

  float s = 0.f;
#pragma unroll
  for (int i = 0; i < 8; ++i)
    s += __expf(d[i][0] - m) + __expf(d[i][1] - m) +
         __expf(d[i][2] - m) + __expf(d[i][3] - m);
#pragma unroll
  for (int off = 16; off > 0; off >>= 1) s += __shfl_xor(s, off, 32);

  const float lse = m + __logf(s);

  const int b   = row / (T_ * U1_);
  const int rem = row - b * (T_ * U1_);
  const int t   = rem / U1_;
  const int u   = rem - t * U1_;

  if (lane == 0) {
    const float* x = (const float*)x4;
    blank_ws[row] = d[0][0] - lse;                  // lp[b,t,u,0]
    if (u < U_) {
      const int lab = labels[b * U_ + u];           // in [1, V)
      const float ev = (u < y_len[b]) ? (x[lab] - lse) : NEGF;
      emit_ws[(b * T_ + t) * U_ + u] = ev;
    }
  }
}

// ---------------------------------------------------------------------------
// Pass 2: alpha DP as an anti-diagonal wavefront, one block per batch.
// blank/emit slab (129KB) staged into LDS via CDNA5 async-to-LDS copies,
// then a single wave walks <=320 diagonals; 65 columns in 3 slots/lane.
// ---------------------------------------------------------------------------
__device__ __forceinline__ float logaddexpf_(float a, float b) {
  const float mx = fmaxf(a, b);
  const float mn = fminf(a, b);
  return mx + __logf(1.0f + __expf(mn - mx));
}

__global__ __launch_bounds__(256)
void rnnt_alpha_kernel(const float* __restrict__ blank_ws,
                       const float* __restrict__ emit_ws,
                       const int* __restrict__ f_len,
                       const int* __restrict__ y_len,
                       float* __restrict__ ll_ws) {
  __shared__ __align__(16) float s_blank[T_ * U1_];   // 66,560 B
  __shared__ __align__(16) float s_emit[T_ * U_];     // 65,536 B
  const int b   = blockIdx.x;
  const int tid = threadIdx.x;

  // ---- async stage: global -> LDS (CDNA5 async tensor path, ASYNCcnt) ----
  {
    const v4f* gb = (const v4f*)(blank_ws + (size_t)b * T_ * U1_);
    const v4f* ge = (const v4f*)(emit_ws + (size_t)b * T_ * U_);
    const uint32_t lb = (uint32_t)(uintptr_t)(&s_blank[0]);
    const uint32_t le = (uint32_t)(uintptr_t)(&s_emit[0]);
    constexpr int NB4 = (T_ * U1_) / 4;   // 4160 float4
    constexpr int NE4 = (T_ * U_) / 4;    // 4096 float4
    for (int i = tid; i < NB4; i += 256) {
      asm volatile("global_load_async_to_lds_b128 %0, %1, off"
                   :: "v"(lb + 16u * (uint32_t)i), "v"(gb + i) : "memory");
    }
    for (int i = tid; i < NE4; i += 256) {
      asm volatile("global_load_async_to_lds_b128 %0, %1, off"
                   :: "v"(le + 16u * (uint32_t)i), "v"(ge + i) : "memory");
    }
    asm volatile("s_wait_asynccnt 0" ::: "memory");
  }
  __syncthreads();
  if (tid >= 32) return;                 // wave 0 runs the DP

  const int lane  = tid;
  const int tlast = f_len[b] - 1;        // >= 127 per setup, >= 0 in general
  const int ustar = y_len[b];            // in [1, U]
  const int dstar = tlast + ustar;

  // prev-diagonal alpha values: lane holds u = lane, lane+32, (lane==0: 64)
  float a0 = NEGF, a1 = NEGF, a2 = NEGF;
  float ll = 0.f;

  const float blank_T = s_blank[tlast * U1_ + ustar];

  for (int d = 0; d <= dstar; ++d) {
    // gather left-neighbor (u-1) values from previous diagonal
    const float p0u  = __shfl_up(a0, 1, 32);
    const float p1u  = __shfl_up(a1, 1, 32);
    const float p031 = __shfl(a0, 31, 32);
    const float p131 = __shfl(a1, 31, 32);
    const float left0 = (lane == 0) ? NEGF : p0u;   // u = lane
    const float left1 = (lane == 0) ? p031 : p1u;   // u = lane+32
    const float left2 = p131;                       // u = 64 (lane 0 only)

    auto cell = [&](int u, float aprev, float lefta) -> float {
      const int t = d - u;
      if (t < 0 || t > tlast) return NEGF;
      if (t == 0 && u == 0) return 0.f;
      const float below = (t >= 1) ? (aprev + s_blank[(t - 1) * U1_ + u]) : NEGF;
      const float left  = (u >= 1) ? (lefta + s_emit[t * U_ + (u - 1)])  : NEGF;
      return logaddexpf_(below, left);
    };

    const float c0 = cell(lane, a0, left0);
    const float c1 = cell(lane + 32, a1, left1);
    const float c2 = (lane == 0) ? cell(64, a2, left2) : NEGF;

    if (d == dstar) {
      if (ustar < 32)      { if (lane == ustar)      ll = c0 + blank_T; }
      else if (ustar < 64) { if (lane == ustar - 32) ll = c1 + blank_T; }
      else                 { if (lane == 0)          ll = c2 + blank_T; }
    }
    a0 = c0; a1 = c1; a2 = c2;
  }

  const int owner = (ustar < 32) ? ustar : ((ustar < 64) ? (ustar - 32) : 0);
  if (lane == owner) ll_ws[b] = ll;
}

// ---------------------------------------------------------------------------
// Pass 3: loss = -mean(ll)
// ---------------------------------------------------------------------------
__global__ __launch_bounds__(32)
void rnnt_finalize_kernel(const float* __restrict__ ll_ws,
                          float* __restrict__ out) {
  const int lane = threadIdx.x;
  float v = (lane < B_) ? ll_ws[lane] : 0.f;
#pragma unroll
  for (int off = 16; off > 0; off >>= 1) v += __shfl_xor(v, off, 32);
  if (lane == 0) out[0] = -v / (float)B_;
}

// ---------------------------------------------------------------------------
extern "C" void kernel_launch(void* const* d_in, const int* in_sizes, int n_in,
                              void* d_out, int out_size, void* d_ws, size_t ws_size,
                              hipStream_t stream) {
  const float* logits = (const float*)d_in[0];
  const int*   labels = (const int*)d_in[1];
  const int*   f_len  = (const int*)d_in[2];
  const int*   y_len  = (const int*)d_in[3];

  float* blank_ws = (float*)d_ws;                       // B*T*(U+1) floats
  float* emit_ws  = blank_ws + (size_t)B_ * T_ * U1_;   // B*T*U floats
  float* ll_ws    = emit_ws  + (size_t)B_ * T_ * U_;    // B floats

  const int rows = B_ * T_ * U1_;                       // 133,120 (multiple of 8)
  rnnt_logprobs_kernel<<<rows / 8, 256, 0, stream>>>(logits, labels, y_len,
                                                     blank_ws, emit_ws);
  rnnt_alpha_kernel<<<B_, 256, 0, stream>>>(blank_ws, emit_ws, f_len, y_len, ll_ws);
  rnnt_finalize_kernel<<<1, 32, 0, stream>>>(ll_ws, (float*)d_out);
}